// PeriodAttention_36739150250184
// MI455X (gfx1250) — compile-verified
//
#include <hip/hip_runtime.h>

// Problem constants (from reference: B=16, L=8192, D=512, PERIOD=64)
#define B_DIM   16
#define L_DIM   8192
#define D_DIM   512
#define PERIOD  64
#define NBLK    128          // L / PERIOD
#define SCALE   0.45f

#define KVSTR   68           // padded LDS row stride for K/V tiles (floats)
#define PSTR    132          // padded LDS row stride for P staging (floats)

typedef __attribute__((ext_vector_type(2))) float v2f;
typedef __attribute__((ext_vector_type(8))) float v8f;

__global__ __launch_bounds__(256) void PeriodAttention_kernel(
    const float* __restrict__ q, const float* __restrict__ k,
    const float* __restrict__ v, float* __restrict__ out,
    float* __restrict__ attn)
{
    extern __shared__ float lds[];
    float* Ks = lds;                       // [128][KVSTR]  key blocks x period
    float* Vs = lds + 128 * KVSTR;         // [128][KVSTR]
    float* Ps = lds + 2 * 128 * KVSTR;     // [8 waves][16][PSTR] softmax staging

    const int bd   = blockIdx.x;           // d fastest -> L2 reuse across d
    const int b    = bd >> 9;              // / D_DIM
    const int d    = bd & (D_DIM - 1);
    const int tid  = threadIdx.x;
    const int lane = tid & 31;
    const int w    = tid >> 5;             // wave id 0..7 -> query strip
    const int half = lane >> 4;            // hi/lo 16-lane half
    const int l16  = lane & 15;

    const size_t in_base = (size_t)b * L_DIM * D_DIM + d;
    const float* qp = q + in_base;
    const float* kp = k + in_base;
    const float* vp = v + in_base;

    // ---- cooperative K/V tile load: element e == sequence position l ----
    for (int e = tid; e < L_DIM; e += 256) {
        const int n = e >> 6, p = e & 63;          // key block, period col
        Ks[n * KVSTR + p] = kp[(size_t)e * D_DIM]; // strided; L2 reuse over d
        Vs[n * KVSTR + p] = vp[(size_t)e * D_DIM];
    }

    // ---- Q strip A-fragments straight to registers (16x64, this wave only)
    // A layout (16x4 f32): lane l: M = l%16, VGPR j: K = 2*(l/16)+j per chunk
    v2f aq[16];
    const int mrow = 16 * w + l16;
    #pragma unroll
    for (int c = 0; c < 16; ++c) {
        const int k0 = 4 * c + 2 * half;
        aq[c].x = qp[(size_t)(mrow * PERIOD + k0)     * D_DIM];
        aq[c].y = qp[(size_t)(mrow * PERIOD + k0 + 1) * D_DIM];
    }

    __syncthreads();

    // ---- S = Q * K^T for strip rows [16w, 16w+16), 8 column tiles ----
    v8f acc[8];
    #pragma unroll
    for (int t = 0; t < 8; ++t) {
        v8f a = {};
        #pragma unroll
        for (int c = 0; c < 16; ++c) {
            const int k0 = 4 * c + 2 * half;
            // B layout: lane l: N(=key block) = l%16, VGPR j: K = 2*half+j
            v2f bf = *(const v2f*)&Ks[(16 * t + l16) * KVSTR + k0];
            a = __builtin_amdgcn_wmma_f32_16x16x4_f32(
                    false, aq[c], false, bf, (short)0, a, false, false);
        }
        acc[t] = a;
    }

    // ---- softmax over key dim (row = r + 8*half + 16w, cols across tiles)
    #pragma unroll
    for (int r = 0; r < 8; ++r) {
        float m = acc[0][r];
        #pragma unroll
        for (int t = 1; t < 8; ++t) m = fmaxf(m, acc[t][r]);
        #pragma unroll
        for (int s = 1; s < 16; s <<= 1) m = fmaxf(m, __shfl_xor(m, s, 32));
        float sum = 0.0f;
        #pragma unroll
        for (int t = 0; t < 8; ++t) {
            const float e = __expf((acc[t][r] - m) * SCALE);
            acc[t][r] = e;
            sum += e;
        }
        #pragma unroll
        for (int s = 1; s < 16; s <<= 1) sum += __shfl_xor(sum, s, 32);
        const float inv = 1.0f / sum;
        #pragma unroll
        for (int t = 0; t < 8; ++t) acc[t][r] *= inv;
    }

    // ---- store attn[b, d, query, key] (second output tensor) ----
    float* attn_bd = attn + (size_t)bd * NBLK * NBLK;
    #pragma unroll
    for (int t = 0; t < 8; ++t) {
        #pragma unroll
        for (int r = 0; r < 8; ++r) {
            attn_bd[(16 * w + r + 8 * half) * NBLK + 16 * t + l16] = acc[t][r];
        }
    }

    // ---- stage P into per-wave private LDS strip (C-layout -> A-layout) ----
    float* Pw = Ps + w * 16 * PSTR;        // [16 local rows][128 keys]
    #pragma unroll
    for (int t = 0; t < 8; ++t) {
        #pragma unroll
        for (int r = 0; r < 8; ++r) {
            Pw[(r + 8 * half) * PSTR + 16 * t + l16] = acc[t][r];
        }
    }
    // per-wave LDS RAW: DS ops are in-order within a wave; no barrier needed.

    // ---- O = P * V : M=16 strip, N=64 (4 tiles), K=128 (32 chunks) ----
    v8f accO[4];
    #pragma unroll
    for (int to = 0; to < 4; ++to) {
        v8f a = {};
        #pragma unroll
        for (int c = 0; c < 32; ++c) {
            const int k0 = 4 * c + 2 * half;
            v2f pa = *(const v2f*)&Pw[l16 * PSTR + k0];
            v2f vb;
            vb.x = Vs[(k0    ) * KVSTR + 16 * to + l16];
            vb.y = Vs[(k0 + 1) * KVSTR + 16 * to + l16];
            a = __builtin_amdgcn_wmma_f32_16x16x4_f32(
                    false, pa, false, vb, (short)0, a, false, false);
        }
        accO[to] = a;
    }

    // ---- scatter O back to [B, L, D] (lines merged in L2 across d) ----
    float* outp = out + in_base;
    #pragma unroll
    for (int to = 0; to < 4; ++to) {
        #pragma unroll
        for (int r = 0; r < 8; ++r) {
            const int qrow = 16 * w + r + 8 * half;
            const int pcol = 16 * to + l16;
            outp[(size_t)(qrow * PERIOD + pcol) * D_DIM] = accO[to][r];
        }
    }
}

extern "C" void kernel_launch(void* const* d_in, const int* in_sizes, int n_in,
                              void* d_out, int out_size, void* d_ws, size_t ws_size,
                              hipStream_t stream) {
    const float* q = (const float*)d_in[0];
    const float* k = (const float*)d_in[1];
    const float* v = (const float*)d_in[2];
    float* out  = (float*)d_out;
    float* attn = out + (size_t)B_DIM * L_DIM * D_DIM;   // outputs concat'd flat

    const size_t shmem = (size_t)(2 * 128 * KVSTR + 8 * 16 * PSTR) * sizeof(float);
    (void)hipFuncSetAttribute((const void*)PeriodAttention_kernel,
                              hipFuncAttributeMaxDynamicSharedMemorySize,
                              (int)shmem);

    PeriodAttention_kernel<<<dim3(B_DIM * D_DIM), dim3(256), shmem, stream>>>(
        q, k, v, out, attn);
}